// LinkPredictor_68650757259726
// MI455X (gfx1250) — compile-verified
//
#include <hip/hip_runtime.h>

// ---------------------------------------------------------------------------
// GCN link predictor for MI455X (gfx1250, wave32).
//   h1 = relu(gcn(x, W1, b1));  z = gcn(h1, W2, b2);  scores = rowdot(z[a], z[b])
// gcn(x,W,b)[i] = dinv[i] * ( sum_{s->i} dinv[s]*(xW)[s] + dinv[i]*(xW)[i] ) + b
// ---------------------------------------------------------------------------

typedef __attribute__((ext_vector_type(2))) float v2f;
typedef __attribute__((ext_vector_type(8))) float v8f;

// ---------------------------------------------------------------------------
// Degree / dinv
// ---------------------------------------------------------------------------
__global__ void deg_kernel(float* __restrict__ deg, const int* __restrict__ dst, int nedges) {
    int e = blockIdx.x * blockDim.x + threadIdx.x;
    if (e < nedges) atomicAdd(&deg[dst[e]], 1.0f);
}

__global__ void finalize_dinv_kernel(float* __restrict__ d, int n) {
    int i = blockIdx.x * blockDim.x + threadIdx.x;
    if (i < n) d[i] = rsqrtf(d[i] + 1.0f);   // +1 = self loop; always > 0
}

// ---------------------------------------------------------------------------
// Tiled fp32 WMMA GEMM:  out rows [row0, row0+16), all NCOLS columns.
// Block = (NCOLS/16) waves; wave w computes the 16x16 tile at columns w*16.
// A tile (16 x K) staged in LDS with +4 padding (bank-conflict-free reads).
// Epilogue: P[row][c] = dinv[row]*acc ; Agg[row][c] = same (self-loop init).
// If FUSE: A element is relu(dinv[row]*rawA + bprev[c])  (layer-1 finalize).
// ---------------------------------------------------------------------------
template <int K, int NCOLS, bool FUSE>
__launch_bounds__(32 * (NCOLS / 16))
__global__ void gcn_gemm_wmma(const float* __restrict__ A,      // [n, K]
                              const float* __restrict__ W,      // [K, NCOLS]
                              const float* __restrict__ bprev,  // [K] (FUSE only)
                              const float* __restrict__ dinv,   // [n]
                              float* __restrict__ P,            // [n, NCOLS]
                              float* __restrict__ Agg,          // [n, NCOLS]
                              int nrows) {
    constexpr int LDK = K + 4;                 // pad: stride%64 == 4 -> conflict-free
    constexpr int NT  = 32 * (NCOLS / 16);
    __shared__ float xs[16 * LDK];

    const int tid  = threadIdx.x;
    const int row0 = blockIdx.x * 16;

    // --- stage 16 x K tile of A into LDS (coalesced along K) ---
    for (int e = tid; e < 16 * K; e += NT) {
        const int r = e / K, c = e % K;
        const int gr = row0 + r;
        float v = 0.0f;
        if (gr < nrows) {
            v = A[(long long)gr * K + c];
            if constexpr (FUSE) v = fmaxf(fmaf(dinv[gr], v, bprev[c]), 0.0f);
        }
        xs[r * LDK + c] = v;
    }
    __syncthreads();

    const int wave = tid >> 5;
    const int lane = tid & 31;
    const int lo   = lane & 15;   // M index (A) / N index (B,D)
    const int half = lane >> 4;   // lane-half selects K pairs
    const int col  = wave * 16 + lo;

    v8f acc = {0.f, 0.f, 0.f, 0.f, 0.f, 0.f, 0.f, 0.f};
#pragma unroll 4
    for (int k = 0; k < K; k += 4) {
        const int ka = k + 2 * half;          // VGPR0/1 hold K=ka, ka+1
        v2f a, b;
        a[0] = xs[lo * LDK + ka];
        a[1] = xs[lo * LDK + ka + 1];
        b[0] = W[ka * NCOLS + col];
        b[1] = W[(ka + 1) * NCOLS + col];
        acc = __builtin_amdgcn_wmma_f32_16x16x4_f32(
            /*neg_a=*/false, a, /*neg_b=*/false, b,
            /*c_mod=*/(short)0, acc, /*reuse_a=*/false, /*reuse_b=*/false);
    }

    // --- epilogue: D layout -> lanes 0-15: M=v ; lanes 16-31: M=v+8 ---
#pragma unroll
    for (int v = 0; v < 8; ++v) {
        const int row = row0 + v + half * 8;
        if (row < nrows) {
            const float val = dinv[row] * acc[v];
            const long long o = (long long)row * NCOLS + col;
            P[o]   = val;   // dinv[s]-scaled message, gathered by scatter
            Agg[o] = val;   // accumulator init == self-loop contribution
        }
    }
}

// ---------------------------------------------------------------------------
// Edge scatter:  one wave per edge;  Agg[dst] += P[src]  (CH channels).
// CH=128 -> float4/lane (b128 coalesced gather), CH=64 -> float2/lane.
// ---------------------------------------------------------------------------
template <int CH>
__global__ void gcn_scatter(const float* __restrict__ P, float* __restrict__ Agg,
                            const int* __restrict__ src, const int* __restrict__ dst,
                            int nedges) {
    const int gid  = blockIdx.x * blockDim.x + threadIdx.x;
    const int e    = gid >> 5;
    const int lane = gid & 31;
    if (e >= nedges) return;
    const int s = src[e], d = dst[e];
    const float* ps = P   + (long long)s * CH;
    float*       ad = Agg + (long long)d * CH;
    if constexpr (CH == 128) {
        const float4 v = reinterpret_cast<const float4*>(ps)[lane];
        const int c = lane * 4;
        atomicAdd(&ad[c + 0], v.x);
        atomicAdd(&ad[c + 1], v.y);
        atomicAdd(&ad[c + 2], v.z);
        atomicAdd(&ad[c + 3], v.w);
    } else {
        const float2 v = reinterpret_cast<const float2*>(ps)[lane];
        const int c = lane * 2;
        atomicAdd(&ad[c + 0], v.x);
        atomicAdd(&ad[c + 1], v.y);
    }
}

// ---------------------------------------------------------------------------
// Score: one wave per edge pair; z[i][c] = dinv[i]*Agg2[i][c] + b2[c] (fused
// layer-2 finalize); 64 channels -> float2 per lane; wave32 xor reduction.
// ---------------------------------------------------------------------------
__global__ void score_kernel(const float* __restrict__ Agg2, const float* __restrict__ dinv,
                             const float* __restrict__ b2, const int* __restrict__ idx,
                             int ep, float* __restrict__ out) {
    const int gid  = blockIdx.x * blockDim.x + threadIdx.x;
    const int e    = gid >> 5;
    const int lane = gid & 31;
    if (e >= ep) return;
    const int a = idx[e], b = idx[ep + e];
    const float da = dinv[a], db = dinv[b];
    const float2 va = reinterpret_cast<const float2*>(Agg2 + (long long)a * 64)[lane];
    const float2 vb = reinterpret_cast<const float2*>(Agg2 + (long long)b * 64)[lane];
    const float2 bb = reinterpret_cast<const float2*>(b2)[lane];
    const float za0 = fmaf(da, va.x, bb.x), za1 = fmaf(da, va.y, bb.y);
    const float zb0 = fmaf(db, vb.x, bb.x), zb1 = fmaf(db, vb.y, bb.y);
    float acc = za0 * zb0 + za1 * zb1;
#pragma unroll
    for (int off = 16; off > 0; off >>= 1) acc += __shfl_xor(acc, off, 32);
    if (lane == 0) out[e] = acc;
}

// ---------------------------------------------------------------------------
// Host launch
// ---------------------------------------------------------------------------
extern "C" void kernel_launch(void* const* d_in, const int* in_sizes, int n_in,
                              void* d_out, int out_size, void* d_ws, size_t ws_size,
                              hipStream_t stream) {
    const float* x    = (const float*)d_in[0];   // [N, 256]
    const int*   edge = (const int*)d_in[1];     // [2, E] (src | dst)
    const int*   pos  = (const int*)d_in[2];     // [2, EP]
    const int*   neg  = (const int*)d_in[3];     // [2, EP]
    const float* W1   = (const float*)d_in[4];   // [256, 128]
    const float* b1   = (const float*)d_in[5];   // [128]
    const float* W2   = (const float*)d_in[6];   // [128, 64]
    const float* b2   = (const float*)d_in[7];   // [64]
    float*       out  = (float*)d_out;           // [2*EP]

    const int n  = in_sizes[0] / 256;  // 100000
    const int e  = in_sizes[1] / 2;    // 1600000
    const int ep = in_sizes[2] / 2;    // 500000

    // workspace carve-out (256B aligned)
    char* ws = (char*)d_ws;
    size_t off = 0;
    auto carve = [&](size_t bytes) -> float* {
        float* p = (float*)(ws + off);
        off += (bytes + 255) & ~(size_t)255;
        return p;
    };
    float* dinv = carve((size_t)n * 4);
    float* P1   = carve((size_t)n * 128 * 4);
    float* Agg1 = carve((size_t)n * 128 * 4);
    float* P2   = carve((size_t)n * 64 * 4);
    float* Agg2 = carve((size_t)n * 64 * 4);

    const int* src = edge;
    const int* dst = edge + e;

    // 1) degrees -> dinv
    hipMemsetAsync(dinv, 0, (size_t)n * 4, stream);
    deg_kernel<<<(e + 255) / 256, 256, 0, stream>>>(dinv, dst, e);
    finalize_dinv_kernel<<<(n + 255) / 256, 256, 0, stream>>>(dinv, n);

    // 2) layer 1: GEMM (WMMA fp32) + scatter
    gcn_gemm_wmma<256, 128, false><<<(n + 15) / 16, 256, 0, stream>>>(
        x, W1, nullptr, dinv, P1, Agg1, n);
    gcn_scatter<128><<<(e * 32 + 255) / 256, 256, 0, stream>>>(P1, Agg1, src, dst, e);

    // 3) layer 2: GEMM with fused relu(dinv*Agg1 + b1) on the A-load
    gcn_gemm_wmma<128, 64, true><<<(n + 15) / 16, 128, 0, stream>>>(
        Agg1, W2, b1, dinv, P2, Agg2, n);
    gcn_scatter<64><<<(e * 32 + 255) / 256, 256, 0, stream>>>(P2, Agg2, src, dst, e);

    // 4) edge scores (layer-2 finalize fused into gathers)
    score_kernel<<<(ep * 32 + 255) / 256, 256, 0, stream>>>(Agg2, dinv, b2, pos, ep, out);
    score_kernel<<<(ep * 32 + 255) / 256, 256, 0, stream>>>(Agg2, dinv, b2, neg, ep, out + ep);
}